// RNNSequenceClassifier_49709951484019
// MI455X (gfx1250) — compile-verified
//
#include <hip/hip_runtime.h>
#include <hip/hip_bf16.h>

// Problem constants (from the reference).
#define BB 64
#define SS 128
#define TT 16
#define E_ELMO 1024
#define D_TRI 100
#define OUT_D (E_ELMO + D_TRI)   // 1124
#define NTOK (BB * SS)           // 8192

typedef __attribute__((ext_vector_type(2))) float v2f;
typedef __attribute__((ext_vector_type(8))) float v8f;

// ---------------------------------------------------------------------------
// Kernel 1: token-embedding gather/copy.  One block per token, 256 threads,
// one float4 per lane (1024 floats = 256 * float4).  Fully coalesced 4 KB
// load + store per block; pure HBM/L2 bandwidth.
// ---------------------------------------------------------------------------
__global__ void __launch_bounds__(256)
tok_copy_kernel(const int* __restrict__ inputs,
                const float* __restrict__ emb,
                float* __restrict__ out) {
  const int token = blockIdx.x;           // 0 .. NTOK-1
  const int t     = threadIdx.x;          // 0 .. 255
  const int row   = inputs[token];
  const float4* src = reinterpret_cast<const float4*>(emb + (size_t)row * E_ELMO);
  float4 v = src[t];
  float* dst = out + (size_t)token * OUT_D;   // OUT_D*4 = 4496, 16B aligned
  reinterpret_cast<float4*>(dst)[t] = v;
}

// ---------------------------------------------------------------------------
// Kernel 2: masked entity-average via V_WMMA_F32_16X16X4_F32, one wave/token.
//   avg[d] = (1/max(cnt,1)) * sum_j m[j] * ent[idx_j][d]
// A[16x4]  = mask chunk broadcast over all M rows (per-lane-half slots)
// B[4x16]  = entity rows for those slots, columns nbase + (lane & 15)
// C[16x16] accumulated over 4 K-chunks; all rows equal, so c[0] on every
// lane is the reduction for its column; lanes 0-15 commit.
//
// No shuffles: slot IDs are compile-time per lane-half, so each lane loads
// its own flags/triples (wave touches ~4 cachelines, L2-resident).  B-gather
// base pointers are hoisted so inner loads are global_load_b32 with
// immediate offsets (nt*64B), coalesced 64B per lane-half.  The D=100 tail
// (tile 6) uses a clamped in-bounds column instead of an EXEC guard, so EXEC
// is all-1s at every WMMA.
// ---------------------------------------------------------------------------
__global__ void __launch_bounds__(128)
triple_avg_kernel(const int* __restrict__ triples,   // [NTOK, TT, 3]
                  const int* __restrict__ flags,     // [NTOK, TT]
                  const float* __restrict__ ent,     // [ENT_VOCAB, D_TRI]
                  float* __restrict__ out) {
  const int lane  = threadIdx.x & 31;
  const int wave  = threadIdx.x >> 5;
  const int token = blockIdx.x * 4 + wave;     // 4 waves / block
  if (token >= NTOK) return;

  const int col    = lane & 15;
  const int kShift = (lane >= 16) ? 2 : 0;

  const int*  tokFlags = flags   + (size_t)token * TT;
  const int*  tokTrip  = triples + (size_t)token * TT * 3;

  // Valid-slot count via wave32 ballot (lanes 0-15 cover the 16 slots).
  const int  fMine  = tokFlags[col];
  const bool vMine  = (fMine == 1) || (fMine == 2);
  const unsigned long long bal = __ballot(vMine && lane < 16);
  const int   cnt  = __popcll(bal);
  const float invd = 1.0f / (float)((cnt > 0) ? cnt : 1);

  // Per-lane A-operand values and B-row base pointers for all 4 K-chunks.
  float        aX[4], aY[4];
  const float* pX[4];
  const float* pY[4];
  size_t       offX[4], offY[4];
  #pragma unroll
  for (int kc = 0; kc < 4; ++kc) {
    const int sX = kc * 4 + kShift;       // slot for a.x / b.x
    const int sY = sX + 1;                // slot for a.y / b.y
    const int fX = tokFlags[sX];
    const int fY = tokFlags[sY];
    const int hX = tokTrip[sX * 3 + 0], tX = tokTrip[sX * 3 + 1];
    const int hY = tokTrip[sY * 3 + 0], tY = tokTrip[sY * 3 + 1];
    aX[kc] = ((fX == 1) || (fX == 2)) ? 1.0f : 0.0f;
    aY[kc] = ((fY == 1) || (fY == 2)) ? 1.0f : 0.0f;
    const int iX = (fX == 1) ? tX : hX;   // reference: where(m1, tail, head)
    const int iY = (fY == 1) ? tY : hY;
    offX[kc] = (size_t)iX * D_TRI;
    offY[kc] = (size_t)iY * D_TRI;
    pX[kc] = ent + offX[kc] + col;
    pY[kc] = ent + offY[kc] + col;
  }

  float* orow = out + (size_t)token * OUT_D + E_ELMO;

  // Tiles 0..5: dims 0..95, no bounds handling needed.
  #pragma unroll
  for (int nt = 0; nt < 6; ++nt) {
    v8f c = {0.f, 0.f, 0.f, 0.f, 0.f, 0.f, 0.f, 0.f};
    #pragma unroll
    for (int kc = 0; kc < 4; ++kc) {
      v2f a; a.x = aX[kc]; a.y = aY[kc];
      v2f b; b.x = pX[kc][nt * 16]; b.y = pY[kc][nt * 16];
      c = __builtin_amdgcn_wmma_f32_16x16x4_f32(
          false, a, false, b, (short)0, c, false, false);
    }
    if (lane < 16) orow[nt * 16 + col] = c[0] * invd;
  }

  // Tile 6: dims 96..111, only 96..99 are real.  Clamp the column so every
  // lane's load stays in-bounds (EXEC stays all-1s); garbage columns are
  // simply never stored.
  {
    const int c6 = (col < D_TRI - 96) ? (96 + col) : (D_TRI - 1);
    v8f c = {0.f, 0.f, 0.f, 0.f, 0.f, 0.f, 0.f, 0.f};
    #pragma unroll
    for (int kc = 0; kc < 4; ++kc) {
      v2f a; a.x = aX[kc]; a.y = aY[kc];
      v2f b; b.x = ent[offX[kc] + c6]; b.y = ent[offY[kc] + c6];
      c = __builtin_amdgcn_wmma_f32_16x16x4_f32(
          false, a, false, b, (short)0, c, false, false);
    }
    if (lane < 16 && col < D_TRI - 96) orow[96 + col] = c[0] * invd;
  }
}

extern "C" void kernel_launch(void* const* d_in, const int* in_sizes, int n_in,
                              void* d_out, int out_size, void* d_ws, size_t ws_size,
                              hipStream_t stream) {
  const int*   inputs  = (const int*)  d_in[0];   // [B,S]           int32
  const int*   triples = (const int*)  d_in[1];   // [B,S,T,3]       int32
  const int*   flags   = (const int*)  d_in[2];   // [B,S,T]         int32
  const float* emb     = (const float*)d_in[3];   // [VOCAB,1024]    f32
  const float* ent     = (const float*)d_in[4];   // [ENT_VOCAB,100] f32
  float*       out     = (float*)d_out;           // [B,S,1124]      f32

  (void)in_sizes; (void)n_in; (void)out_size; (void)d_ws; (void)ws_size;

  // Part 1: token embedding rows (cols 0..1023).
  tok_copy_kernel<<<NTOK, 256, 0, stream>>>(inputs, emb, out);

  // Part 2: masked entity average (cols 1024..1123), one wave per token.
  triple_avg_kernel<<<NTOK / 4, 128, 0, stream>>>(triples, flags, ent, out);
}